// LCAWSupConLoss_57664230916706
// MI455X (gfx1250) — compile-verified
//
#include <hip/hip_runtime.h>
#include <hip/hip_bf16.h>
#include <math.h>

typedef float v2f __attribute__((ext_vector_type(2)));
typedef float v8f __attribute__((ext_vector_type(8)));

#define NB 4096
#define DD 128
#define EPSN 1e-12f
#define EPSP 1e-5f
#define CSPLIT 16                    // column-range split per 16-row strip
#define COLS_PER_CHUNK (NB / CSPLIT) // 256 columns -> 16 tiles per wave

// ---------------- Kernel 1: row L2-normalize (one wave32 per row) ----------------
__global__ void __launch_bounds__(256) lca_normalize(const float* __restrict__ x,
                                                     float* __restrict__ fn) {
  const int wave = (blockIdx.x * blockDim.x + threadIdx.x) >> 5;  // row id
  const int lane = threadIdx.x & 31;
  if (wave >= NB) return;
  const float4 v = ((const float4*)(x + wave * DD))[lane];
  float ss = v.x * v.x + v.y * v.y + v.z * v.z + v.w * v.w;
#pragma unroll
  for (int off = 16; off >= 1; off >>= 1) ss += __shfl_xor(ss, off, 32);
  const float scale = 1.0f / fmaxf(sqrtf(ss), EPSN);
  float4 o;
  o.x = v.x * scale; o.y = v.y * scale; o.z = v.z * scale; o.w = v.w * scale;
  ((float4*)(fn + wave * DD))[lane] = o;
}

// ---------------- Kernel 2: fused sim-GEMM + masked softmax partials ----------------
// One wave per (16-row strip, column chunk). For each 16-col tile: 16x16x128 f32
// WMMA, then per-element accumulation of sum(exp(v-1)), pos_sum, pos_cnt.
// Partials written per (row, chunk); merged deterministically in lca_finalize.
__global__ void __launch_bounds__(128) lca_supcon(const float* __restrict__ fn,
                                                  const long long* __restrict__ labels,
                                                  float* __restrict__ spart,
                                                  float* __restrict__ ppart,
                                                  float* __restrict__ cpart) {
  const int wave_in_block = threadIdx.x >> 5;
  const int lane = threadIdx.x & 31;
  const int strip = ((blockIdx.x & 63) << 2) + wave_in_block;  // 0..255
  const int chunk = blockIdx.x >> 6;                           // 0..CSPLIT-1
  const int row_base = strip << 4;
  const int col_base = chunk * COLS_PER_CHUNK;
  const int l15 = lane & 15;
  const int hi  = lane >> 4;

  // A fragment for V_WMMA_F32_16X16X4_F32:
  // lane l, vgpr v  ->  A[M = l&15][K = 2*(l>>4) + v]; full K=128 => 32 frags.
  const float* arow = fn + (row_base + l15) * DD + 2 * hi;
  v2f a[32];
#pragma unroll
  for (int k = 0; k < 32; ++k) { a[k].x = arow[4 * k]; a[k].y = arow[4 * k + 1]; }

  // Labels of the 8 rows this lane accumulates (C layout: M = r + 8*hi).
  int labm[8];
#pragma unroll
  for (int r = 0; r < 8; ++r) labm[r] = (int)labels[row_base + r + 8 * hi];

  float srow[8], psum[8], pcnt[8];
#pragma unroll
  for (int r = 0; r < 8; ++r) { srow[r] = 0.f; psum[r] = 0.f; pcnt[r] = 0.f; }

  for (int col = col_base; col < col_base + COLS_PER_CHUNK; col += 16) {
    // B fragment: lane l, vgpr v -> B[K = 2*hi + v][N = l&15] = fn[col + (l&15)][...]
    const float* brow = fn + (col + l15) * DD + 2 * hi;
    const int labn = (int)labels[col + l15];
    v8f c0 = {};
    v8f c1 = {};
#pragma unroll
    for (int k = 0; k < 32; k += 2) {
      v2f b0; b0.x = brow[4 * k];       b0.y = brow[4 * k + 1];
      v2f b1; b1.x = brow[4 * (k + 1)]; b1.y = brow[4 * (k + 1) + 1];
      c0 = __builtin_amdgcn_wmma_f32_16x16x4_f32(false, a[k],     false, b0,
                                                 (short)0, c0, false, false);
      c1 = __builtin_amdgcn_wmma_f32_16x16x4_f32(false, a[k + 1], false, b1,
                                                 (short)0, c1, false, false);
    }
    const int n = col + l15;
#pragma unroll
    for (int r = 0; r < 8; ++r) {
      const int m_idx = row_base + r + 8 * hi;
      if (n == m_idx) continue;                 // diagonal: excluded everywhere
      const float v = c0[r] + c1[r];            // sim in [-1,1] (tau = 1)
      srow[r] += __expf(v - 1.0f);              // fixed shift: max(sim) <= 1
      if (labn == labm[r]) { psum[r] += v; pcnt[r] += 1.0f; }
    }
  }

  // Reduce across the 16 lanes of each half-wave sharing the same rows.
#pragma unroll
  for (int off = 1; off < 16; off <<= 1) {
#pragma unroll
    for (int r = 0; r < 8; ++r) {
      srow[r] += __shfl_xor(srow[r], off, 32);
      psum[r] += __shfl_xor(psum[r], off, 32);
      pcnt[r] += __shfl_xor(pcnt[r], off, 32);
    }
  }

  if (l15 == 0) {  // lanes 0 and 16 hold rows [row_base+8*hi .. +7]
#pragma unroll
    for (int r = 0; r < 8; ++r) {
      const int idx = (row_base + r + 8 * hi) * CSPLIT + chunk;
      spart[idx] = srow[r];
      ppart[idx] = psum[r];
      cpart[idx] = pcnt[r];
    }
  }
}

// ---------------- Kernel 3: deterministic merge + final reduction ----------------
__global__ void __launch_bounds__(256) lca_finalize(const float* __restrict__ spart,
                                                    const float* __restrict__ ppart,
                                                    const float* __restrict__ cpart,
                                                    float* __restrict__ out) {
  __shared__ float sh[256];
  float acc = 0.f;
  for (int row = threadIdx.x; row < NB; row += 256) {
    float s = 0.f, p = 0.f, c = 0.f;
#pragma unroll
    for (int k = 0; k < CSPLIT; ++k) {
      s += spart[row * CSPLIT + k];
      p += ppart[row * CSPLIT + k];
      c += cpart[row * CSPLIT + k];
    }
    const float lse = 1.0f + __logf(s);
    acc += -(p - c * lse) / (c + EPSP);
  }
  sh[threadIdx.x] = acc;
  __syncthreads();
  for (int off = 128; off >= 1; off >>= 1) {
    if ((int)threadIdx.x < off) sh[threadIdx.x] += sh[threadIdx.x + off];
    __syncthreads();
  }
  if (threadIdx.x == 0) out[0] = sh[0] / (float)NB;
}

extern "C" void kernel_launch(void* const* d_in, const int* in_sizes, int n_in,
                              void* d_out, int out_size, void* d_ws, size_t ws_size,
                              hipStream_t stream) {
  (void)in_sizes; (void)n_in; (void)out_size; (void)ws_size;
  const float* feat = (const float*)d_in[0];
  const long long* labels = (const long long*)d_in[1];
  float* out = (float*)d_out;

  float* fn    = (float*)d_ws;               // 4096*128 f32 = 2 MB (L2-resident)
  float* spart = fn + NB * DD;               // NB*CSPLIT f32
  float* ppart = spart + NB * CSPLIT;        // NB*CSPLIT f32
  float* cpart = ppart + NB * CSPLIT;        // NB*CSPLIT f32

  lca_normalize<<<NB / 8, 256, 0, stream>>>(feat, fn);          // 8 waves/block
  lca_supcon<<<64 * CSPLIT, 128, 0, stream>>>(fn, labels, spart, ppart, cpart);
  lca_finalize<<<1, 256, 0, stream>>>(spart, ppart, cpart, out);
}